// PatchConsistencyLoss_54666343744090
// MI455X (gfx1250) — compile-verified
//
#include <hip/hip_runtime.h>
#include <hip/hip_bf16.h>
#include <stdint.h>

typedef float v2f __attribute__((ext_vector_type(2)));
typedef float v8f __attribute__((ext_vector_type(8)));

#define VOCAB 3717
#define NPATCH 16384   // 32 * (8*8*8)
#define AIR0 102
#define AIR1 576
#define AIR2 3352

__device__ __forceinline__ int is_air(int t) {
    return (t == AIR0) | (t == AIR1) | (t == AIR2);
}

// One wave == one patch. Lanes 0-15 async-DMA one 16B row (4 int32 tokens)
// each into LDS; then lane l evaluates tokens l and l+32.
__global__ void patch_entropy_kernel(const int* __restrict__ structure,
                                     float* __restrict__ patch_ent) {
    __shared__ int tokens[8][64];   // 8 waves/block * 64 tokens * 4B = 2KB LDS

    const int lane  = threadIdx.x & 31;
    const int wslot = threadIdx.x >> 5;
    const int p     = blockIdx.x * 8 + wslot;     // patch id 0..16383
    if (p >= NPATCH) return;

    // patch coordinates: p = ((b*8 + pd)*8 + ph)*8 + pw
    const int b  = p >> 9;
    const int r  = p & 511;
    const int z0 = ((r >> 6) & 7) << 2;
    const int y0 = ((r >> 3) & 7) << 2;
    const int x0 = (r & 7) << 2;

    // --- async global -> LDS: 16 rows of 16 bytes, lanes 0..15 ---
    if (lane < 16) {
        const int zr = lane >> 2;          // 0..3
        const int yr = lane & 3;           // 0..3
        const long long elem =
            (((long long)(b * 32 + z0 + zr) * 32 + (y0 + yr)) * 32 + x0);
        unsigned long long gaddr = (unsigned long long)(structure + elem);
        unsigned lds_addr = (unsigned)(uintptr_t)(&tokens[wslot][lane * 4]);
        asm volatile("global_load_async_to_lds_b128 %0, %1, off"
                     :: "v"(lds_addr), "v"(gaddr) : "memory");
    }
    asm volatile("s_wait_asynccnt 0" ::: "memory");

    const int* tok = tokens[wslot];
    const int t0 = tok[lane];        // per-lane distinct, stride-4B: conflict-free
    const int t1 = tok[lane + 32];

    // 64x64 pairwise equality counts + non-air total (uniform broadcast reads)
    int c0 = 0, c1 = 0, T = 0;
    const int4* q = (const int4*)tok;
#pragma unroll
    for (int j = 0; j < 16; ++j) {
        int4 u = q[j];                // same address in all lanes -> broadcast
        c0 += (t0 == u.x) + (t0 == u.y) + (t0 == u.z) + (t0 == u.w);
        c1 += (t1 == u.x) + (t1 == u.y) + (t1 == u.z) + (t1 == u.w);
        T  += (!is_air(u.x)) + (!is_air(u.y)) + (!is_air(u.z)) + (!is_air(u.w));
    }

    // entropy = -(1/T) * sum_{non-air i} log(c_i/T + 1e-10); all-air -> 0
    const float invT = (T > 0) ? (1.0f / (float)T) : 0.0f;
    float e = 0.0f;
    if (T > 0) {
        if (!is_air(t0)) e += __logf((float)c0 * invT + 1e-10f);
        if (!is_air(t1)) e += __logf((float)c1 * invT + 1e-10f);
    }
    // wave32 butterfly reduction
    for (int off = 16; off > 0; off >>= 1)
        e += __shfl_xor(e, off, 32);

    if (lane == 0) patch_ent[p] = -invT * e;
}

// Deterministic 16384 -> 1 reduction using v_wmma_f32_16x16x4_f32:
// D += A(16x4 chunk of values) * B(4x16 ones). Row-sums accumulate the chunk.
__global__ void reduce_entropy_kernel(const float* __restrict__ patch_ent,
                                      float* __restrict__ out) {
    __shared__ float partial[8];
    const int lane = threadIdx.x & 31;
    const int w    = threadIdx.x >> 5;

    const float* chunk = patch_ent + w * 2048;   // 8 waves * 2048 = 16384
    // A layout (16x4 f32): lanes 0-15 hold A[m][0..1], lanes 16-31 hold A[m][2..3]
    const int idx = (lane & 15) * 4 + ((lane >> 4) << 1);

    v8f c = {0.f, 0.f, 0.f, 0.f, 0.f, 0.f, 0.f, 0.f};
    v2f bones; bones.x = 1.0f; bones.y = 1.0f;   // B(4x16) all-ones

    for (int it = 0; it < 32; ++it) {            // 32 * 64 = 2048 values
        const float* pb = chunk + it * 64;
        v2f a; a.x = pb[idx]; a.y = pb[idx + 1];
        c = __builtin_amdgcn_wmma_f32_16x16x4_f32(
                false, a, false, bones, (short)0, c, false, false);
    }
    // lanes 0-15: sum of rows 0-7; lanes 16-31: rows 8-15 (all columns equal)
    float s = c[0] + c[1] + c[2] + c[3] + c[4] + c[5] + c[6] + c[7];
    s += __shfl_xor(s, 16, 32);
    if (lane == 0) partial[w] = s;
    __syncthreads();
    if (threadIdx.x == 0) {
        float tot = 0.0f;
        for (int i = 0; i < 8; ++i) tot += partial[i];
        out[0] = tot * (1.0f / (16384.0f + 1e-6f));
    }
}

extern "C" void kernel_launch(void* const* d_in, const int* in_sizes, int n_in,
                              void* d_out, int out_size, void* d_ws, size_t ws_size,
                              hipStream_t stream) {
    (void)in_sizes; (void)n_in; (void)out_size; (void)ws_size;
    const int* structure = (const int*)d_in[0];
    float* patch_ent = (float*)d_ws;         // 16384 floats = 64 KB scratch
    float* out = (float*)d_out;

    patch_entropy_kernel<<<NPATCH / 8, 256, 0, stream>>>(structure, patch_ent);
    reduce_entropy_kernel<<<1, 256, 0, stream>>>(patch_ent, out);
}